// HilbertTransformWithNoise_86303072846442
// MI455X (gfx1250) — compile-verified
//
#include <hip/hip_runtime.h>
#include <hip/hip_bf16.h>

// Problem constants (from reference): B=8, L=8192, C=512, POOL_K=3
#define BB 8
#define LL 8192
#define CC 512
#define ROWS (BB * LL)              // 65536 rows of noise
#define OUT_PER_B (LL - 2)          // 8190
#define OUT_TOTAL (BB * OUT_PER_B)  // 65520

typedef __attribute__((ext_vector_type(2))) float v2f;
typedef __attribute__((ext_vector_type(4))) float v4f;
typedef __attribute__((ext_vector_type(8))) float v8f;

static __device__ __forceinline__ float sum4(const v4f v) {
    return (v[0] + v[1]) + (v[2] + v[3]);
}

// ---------------------------------------------------------------------------
// Kernel 1: per-block partial sums of x.  1024 blocks x 256 threads.
// Each block owns a contiguous 32768-float chunk (8192 float4), fully inside
// one batch (4M floats per batch = 128 blocks/batch).  Deterministic LDS tree.
// Loads chunked 8-deep to keep VGPR pressure low (occupancy >> latency).
// ---------------------------------------------------------------------------
__global__ void __launch_bounds__(256) x_partial_kernel(const float* __restrict__ x,
                                                        float* __restrict__ partials) {
    const int t = threadIdx.x;
    const v4f* x4 = (const v4f*)x + (long)blockIdx.x * 8192 + t;
    float acc = 0.f;
#pragma unroll 1
    for (int i = 0; i < 32; i += 8) {
        float s = 0.f;
#pragma unroll
        for (int j = 0; j < 8; ++j)
            s += sum4(__builtin_nontemporal_load(&x4[(i + j) * 256]));
        acc += s;
    }
    __shared__ float sm[256];
    sm[t] = acc;
    __syncthreads();
#pragma unroll
    for (int s = 128; s > 0; s >>= 1) {
        if (t < s) sm[t] += sm[t + s];
        __syncthreads();
    }
    if (t == 0) partials[blockIdx.x] = sm[0];
}

// ---------------------------------------------------------------------------
// Kernel 2: finalize per-batch mean M[b] = (sum over 128 partials) / (L*C).
// ---------------------------------------------------------------------------
__global__ void m_final_kernel(const float* __restrict__ partials,
                               float* __restrict__ M) {
    const int b = threadIdx.x;
    if (b < BB) {
        float s = 0.f;
        for (int i = 0; i < 128; ++i) s += partials[b * 128 + i];
        M[b] = s / (float)(LL * CC);
    }
}

// ---------------------------------------------------------------------------
// Kernel 3: row sums of noise via V_WMMA_F32_16X16X4_F32.
// Block = 256 threads = 8 waves; wave handles 16 rows (block: 128 rows),
// grid = 512 blocks -> 65536 rows.
// Lane layout matches the 32-bit 16x4 A-matrix spec:
//   lane l<16 : M=l, K=0 (VGPR0), K=1 (VGPR1)   -> channel blocks 0,1
//   lane l>=16: M=l-16, K=2 (VGPR0), K=3 (VGPR1) -> channel blocks 2,3
// Each lane accumulates two 128-channel partials from a contiguous 1KiB run
// of its row (every fetched line fully consumed).  B = ones, so
// D[m,n] = sum_k A[m,k] = full 512-channel row sum, replicated over N.
// Loads chunked 8-deep per iteration to keep VGPR usage modest.
// ---------------------------------------------------------------------------
__global__ void __launch_bounds__(256) rowsum_wmma_kernel(const float* __restrict__ noise,
                                                          float* __restrict__ S) {
    const int lane = threadIdx.x & 31;
    const int wave = threadIdx.x >> 5;
    const long rowbase = (long)blockIdx.x * 128 + wave * 16;
    const int m  = lane & 15;   // row within the wave's 16-row tile
    const int hi = lane >> 4;   // 0 -> blocks {0,1}; 1 -> blocks {2,3}

    const v4f* rp = (const v4f*)(noise + (rowbase + m) * CC + hi * 256);

    float p0 = 0.f, p1 = 0.f;
#pragma unroll 1
    for (int i = 0; i < 32; i += 8) {      // channels [hi*256, hi*256+128)
        float s = 0.f;
#pragma unroll
        for (int j = 0; j < 8; ++j)
            s += sum4(__builtin_nontemporal_load(&rp[i + j]));
        p0 += s;
    }
#pragma unroll 1
    for (int i = 32; i < 64; i += 8) {     // channels [hi*256+128, hi*256+256)
        float s = 0.f;
#pragma unroll
        for (int j = 0; j < 8; ++j)
            s += sum4(__builtin_nontemporal_load(&rp[i + j]));
        p1 += s;
    }

    v2f a; a[0] = p0; a[1] = p1;           // A[m,k] = partial(row m, block k)
    v2f b; b[0] = 1.f; b[1] = 1.f;         // B = ones -> row-sum reduction
    v8f c = {};
    c = __builtin_amdgcn_wmma_f32_16x16x4_f32(
            /*neg_a=*/false, a, /*neg_b=*/false, b,
            /*c_mod=*/(short)0, c, /*reuse_a=*/false, /*reuse_b=*/false);

    // D layout: VGPR r -> M=r (lanes 0-15), M=r+8 (lanes 16-31); replicated in N.
    if (lane == 0) {
#pragma unroll
        for (int r = 0; r < 8; ++r) S[rowbase + r] = c[r];
    } else if (lane == 16) {
#pragma unroll
        for (int r = 0; r < 8; ++r) S[rowbase + 8 + r] = c[r];
    }
}

// ---------------------------------------------------------------------------
// Kernel 4: pooled output.
// out[b,l] = M[b] + (0.1/(3*C)) * (S[b,l] + S[b,l+1] + S[b,l+2])
// ---------------------------------------------------------------------------
__global__ void pool_kernel(const float* __restrict__ M,
                            const float* __restrict__ S,
                            float* __restrict__ out) {
    int idx = blockIdx.x * 256 + threadIdx.x;
    if (idx >= OUT_TOTAL) return;
    int b = idx / OUT_PER_B;
    int l = idx - b * OUT_PER_B;
    const float* Sb = S + (long)b * LL + l;
    const float k = 0.1f / (3.0f * (float)CC);
    out[idx] = M[b] + k * ((Sb[0] + Sb[1]) + Sb[2]);
}

// ---------------------------------------------------------------------------
// Host launcher
// ---------------------------------------------------------------------------
extern "C" void kernel_launch(void* const* d_in, const int* in_sizes, int n_in,
                              void* d_out, int out_size, void* d_ws, size_t ws_size,
                              hipStream_t stream) {
    (void)in_sizes; (void)n_in; (void)out_size; (void)ws_size;
    const float* x     = (const float*)d_in[0];
    const float* noise = (const float*)d_in[1];
    float* out = (float*)d_out;

    float* ws       = (float*)d_ws;
    float* partials = ws;            // [1024]
    float* M        = ws + 1024;     // [8]
    float* S        = ws + 2048;     // [65536]

    x_partial_kernel<<<1024, 256, 0, stream>>>(x, partials);
    m_final_kernel<<<1, 32, 0, stream>>>(partials, M);
    rowsum_wmma_kernel<<<ROWS / 128, 256, 0, stream>>>(noise, S);
    pool_kernel<<<(OUT_TOTAL + 255) / 256, 256, 0, stream>>>(M, S, out);
}